// GNNDecoder_79310866088343
// MI455X (gfx1250) — compile-verified
//
#include <hip/hip_runtime.h>

typedef __attribute__((ext_vector_type(2))) float v2f;
typedef __attribute__((ext_vector_type(8))) float v8f;
typedef __attribute__((ext_vector_type(4))) unsigned int u32x4;
typedef __attribute__((ext_vector_type(4))) int i32x4;
typedef __attribute__((ext_vector_type(8))) int i32x8;

#define TILE_ROWS 128
#define D_DIM 128
#define DFF_DIM 256
#define OUT_DIM 119

#if defined(__AMDGCN__)
#define ON_DEVICE 1
#else
#define ON_DEVICE 0
#endif

#if ON_DEVICE && __has_builtin(__builtin_amdgcn_tensor_load_to_lds) && \
    __has_builtin(__builtin_amdgcn_s_wait_tensorcnt)
#define HAVE_TDM 1
#else
#define HAVE_TDM 0
#endif

// ---------------------------------------------------------------------------
// CDNA5 async global->LDS copy (16B per lane). Dynamic LDS base is offset 0,
// so LDS byte offsets are computed arithmetically.
// ---------------------------------------------------------------------------
__device__ __forceinline__ void async_g2l_b128(const void* gptr, unsigned lds_byte_off) {
#if ON_DEVICE
  asm volatile("global_load_async_to_lds_b128 %0, %1, off"
               :: "v"(lds_byte_off), "v"(gptr)
               : "memory");
#endif
}
__device__ __forceinline__ void wait_async0() {
#if ON_DEVICE
  asm volatile("s_wait_asynccnt 0" ::: "memory");
#endif
}

// ---------------------------------------------------------------------------
// TDM: one-instruction 2D tile DMA global->LDS with hardware row padding.
// pad_interval code: 6 => every 128 dwords, 7 => every 256 dwords.
// pad_amount  code: 3 => 4 dwords  (gives LDS row strides 132 / 260).
// Descriptor layout per CDNA5 ISA section 8.3/8.4 (D# groups 0 and 1).
// This toolchain's builtin takes 6 args (g0, g1, g2, g3, extra group, cpol).
// ---------------------------------------------------------------------------
#if HAVE_TDM
__device__ __forceinline__ void tdm_load_2d(const void* gptr, unsigned lds_byte_addr,
                                            unsigned width, unsigned rows,
                                            unsigned row_stride,
                                            unsigned pad_interval_code,
                                            unsigned pad_amount_code) {
  const unsigned long long ga = (unsigned long long)gptr;
  u32x4 g0;
  g0[0] = 1u;                                            // count=1, user descriptor
  g0[1] = lds_byte_addr;                                 // lds_addr [63:32]
  g0[2] = (unsigned)(ga & 0xffffffffu);                  // global_addr lo
  g0[3] = (unsigned)((ga >> 32) & 0x01ffffffu) | (2u << 30);  // addr hi | type=2
  i32x8 g1;
  g1[0] = (int)((2u << 16) |                             // data_size = 4B
                (1u << 20) |                             // pad_enable
                (pad_interval_code << 22) |
                (pad_amount_code << 25));
  g1[1] = (int)(width << 16);                            // tensor_dim0 [79:48]
  g1[2] = (int)(rows << 16);                             // tensor_dim1 [111:80]
  g1[3] = (int)(width << 16);                            // tile_dim0   [127:112]
  g1[4] = (int)rows;                                     // tile_dim1   [143:128]
  g1[5] = (int)row_stride;                               // tensor_dim0_stride lo
  g1[6] = 0;                                             // stride hi / dim1_stride lo
  g1[7] = 0;
  i32x4 z4 = {0, 0, 0, 0};
  i32x8 z8 = {0, 0, 0, 0, 0, 0, 0, 0};
  __builtin_amdgcn_tensor_load_to_lds(g0, g1, z4, z4, z8, 0);
}
#endif

// ---------------------------------------------------------------------------
// Kernel 1: h = prelu(x) @ W_enc^T   (fp32 WMMA 16x16x4)
// LDS: As[128][132], Bs[128][132]  (stride 132 == 4 mod 64 dwords -> no bank conflicts)
// ---------------------------------------------------------------------------
__global__ __launch_bounds__(256) void encode_kernel(
    const float* __restrict__ x, const float* __restrict__ Wenc,
    const float* __restrict__ prelu_a, float* __restrict__ h, int n) {
  extern __shared__ float lds[];
  float* As = lds;                 // [128][132]
  float* Bs = lds + 128 * 132;     // [128][132]
  const int tid = threadIdx.x;
  const int r0 = blockIdx.x * TILE_ROWS;
  const float pa = *prelu_a;

  // Stage W_enc into Bs (B[k][n] = W_enc[n][k] => direct row copy w/ padding)
#if HAVE_TDM
  if (tid < 32) {
    tdm_load_2d(Wenc, (unsigned)(128 * 132 * 4), 128, 128, 128, 6, 3);
  }
#else
  for (int i = tid; i < 128 * 32; i += 256) {
    const int row = i >> 5;
    const int c4 = (i & 31) << 2;
    const float4 v = *(const float4*)(Wenc + row * 128 + c4);
    float* dst = Bs + row * 132 + c4;
    dst[0] = v.x; dst[1] = v.y; dst[2] = v.z; dst[3] = v.w;
  }
#endif
  // Stage prelu(x) tile (prelu fused on the way in; zero-fill past n)
  for (int i = tid; i < 128 * 32; i += 256) {
    const int row = i >> 5;
    const int c4 = (i & 31) << 2;
    const int gr = r0 + row;
    float4 v = make_float4(0.f, 0.f, 0.f, 0.f);
    if (gr < n) v = *(const float4*)(x + (size_t)gr * D_DIM + c4);
    float* dst = As + row * 132 + c4;
    dst[0] = v.x > 0.f ? v.x : pa * v.x;
    dst[1] = v.y > 0.f ? v.y : pa * v.y;
    dst[2] = v.z > 0.f ? v.z : pa * v.z;
    dst[3] = v.w > 0.f ? v.w : pa * v.w;
  }
#if HAVE_TDM
  if (tid < 32) __builtin_amdgcn_s_wait_tensorcnt(0);
#endif
  __syncthreads();

  const int wv = tid >> 5;
  const int lane = tid & 31;
  const int half = lane >> 4;
  const int ln = lane & 15;
  const float* ap = As + (wv * 16 + ln) * 132 + 2 * half;

  for (int tn = 0; tn < 8; tn += 2) {
    const float* bp0 = Bs + (tn * 16 + ln) * 132 + 2 * half;
    const float* bp1 = Bs + ((tn + 1) * 16 + ln) * 132 + 2 * half;
    v8f c0 = {}, c1 = {};
#pragma unroll
    for (int k = 0; k < 32; ++k) {
      const v2f av = *(const v2f*)(ap + 4 * k);
      const v2f b0 = *(const v2f*)(bp0 + 4 * k);
      const v2f b1v = *(const v2f*)(bp1 + 4 * k);
      c0 = __builtin_amdgcn_wmma_f32_16x16x4_f32(false, av, false, b0,
                                                 (short)0, c0, false, false);
      c1 = __builtin_amdgcn_wmma_f32_16x16x4_f32(false, av, false, b1v,
                                                 (short)0, c1, false, false);
    }
    const int col0 = tn * 16 + ln;
    const int col1 = col0 + 16;
#pragma unroll
    for (int v = 0; v < 8; ++v) {
      const int row = r0 + wv * 16 + v + 8 * half;
      if (row < n) {
        h[(size_t)row * D_DIM + col0] = c0[v];
        h[(size_t)row * D_DIM + col1] = c1[v];
      }
    }
  }
}

// ---------------------------------------------------------------------------
// Kernel 2: zero masked rows of h
// ---------------------------------------------------------------------------
__global__ __launch_bounds__(256) void mask_kernel(
    const int* __restrict__ masked_idx, float* __restrict__ h, int nmask) {
  const int i = blockIdx.x * blockDim.x + threadIdx.x;  // nmask * 32 lanes
  if (i >= nmask * 32) return;
  const int r = masked_idx[i >> 5];
  const int c4 = (i & 31) << 2;
  *(float4*)(h + (size_t)r * D_DIM + c4) = make_float4(0.f, 0.f, 0.f, 0.f);
}

// ---------------------------------------------------------------------------
// Kernel 3: aggr[i] = h[i] + emb1[4] + emb2[0]   (self-loop init)
// ---------------------------------------------------------------------------
__global__ __launch_bounds__(256) void selfloop_kernel(
    const float* __restrict__ h, const float* __restrict__ emb1,
    const float* __restrict__ emb2, float* __restrict__ aggr, int n) {
  const int i = blockIdx.x * blockDim.x + threadIdx.x;
  if (i >= n * 32) return;
  const int row = i >> 5;
  const int c4 = (i & 31) << 2;
  const float4 hv = *(const float4*)(h + (size_t)row * D_DIM + c4);
  const float4 e1 = *(const float4*)(emb1 + 4 * D_DIM + c4);
  const float4 e2 = *(const float4*)(emb2 + c4);
  float4 o;
  o.x = hv.x + e1.x + e2.x;
  o.y = hv.y + e1.y + e2.y;
  o.z = hv.z + e1.z + e2.z;
  o.w = hv.w + e1.w + e2.w;
  *(float4*)(aggr + (size_t)row * D_DIM + c4) = o;
}

// ---------------------------------------------------------------------------
// Kernel 4: edge scatter — wave per edge, lane handles 4 columns,
// fp32 global atomic adds into aggr[dst]
// ---------------------------------------------------------------------------
__global__ __launch_bounds__(256) void scatter_kernel(
    const float* __restrict__ h, const int* __restrict__ edge_index,
    const int* __restrict__ edge_attr, const float* __restrict__ emb1,
    const float* __restrict__ emb2, float* __restrict__ aggr, int ne) {
  const int gid = blockIdx.x * blockDim.x + threadIdx.x;
  const int e = gid >> 5;
  if (e >= ne) return;
  const int lane = gid & 31;
  const int s = edge_index[e];
  const int d = edge_index[ne + e];
  const int et = edge_attr[2 * e];
  const int ed = edge_attr[2 * e + 1];
  const int c4 = lane << 2;
  const float4 hv = *(const float4*)(h + (size_t)s * D_DIM + c4);
  const float4 e1 = *(const float4*)(emb1 + et * D_DIM + c4);
  const float4 e2 = *(const float4*)(emb2 + ed * D_DIM + c4);
  float* ap = aggr + (size_t)d * D_DIM + c4;
  atomicAdd(ap + 0, hv.x + e1.x + e2.x);
  atomicAdd(ap + 1, hv.y + e1.y + e2.y);
  atomicAdd(ap + 2, hv.z + e1.z + e2.z);
  atomicAdd(ap + 3, hv.w + e1.w + e2.w);
}

// ---------------------------------------------------------------------------
// Kernel 5: hid = relu(aggr @ W1^T + b1)    K=128, 256 output cols
// LDS: As[128][132] + Bs[256][132]
// ---------------------------------------------------------------------------
__global__ __launch_bounds__(256) void mlp1_kernel(
    const float* __restrict__ aggr, const float* __restrict__ W1,
    const float* __restrict__ b1, float* __restrict__ hid, int n) {
  extern __shared__ float lds[];
  float* As = lds;                 // [128][132]
  float* Bs = lds + 128 * 132;     // [256][132]
  const int tid = threadIdx.x;
  const int r0 = blockIdx.x * TILE_ROWS;

  // W1 -> Bs via TDM (fallback: staged copy)
#if HAVE_TDM
  if (tid < 32) {
    tdm_load_2d(W1, (unsigned)(128 * 132 * 4), 128, 256, 128, 6, 3);
  }
#else
  for (int i = tid; i < 256 * 32; i += 256) {
    const int row = i >> 5;
    const int c4 = (i & 31) << 2;
    const float4 v = *(const float4*)(W1 + row * D_DIM + c4);
    float* dst = Bs + row * 132 + c4;
    dst[0] = v.x; dst[1] = v.y; dst[2] = v.z; dst[3] = v.w;
  }
#endif
  // aggr tile -> As via per-lane async b128 copies (no transform needed)
#if ON_DEVICE
  for (int i = tid; i < 128 * 32; i += 256) {
    const int row = i >> 5;
    const int c4 = (i & 31) << 2;
    const int gr = r0 + row;
    const unsigned loff = (unsigned)((row * 132 + c4) * 4);
    if (gr < n) {
      async_g2l_b128(aggr + (size_t)gr * D_DIM + c4, loff);
    } else {
      *(float4*)(As + row * 132 + c4) = make_float4(0.f, 0.f, 0.f, 0.f);
    }
  }
  wait_async0();
#else
  for (int i = tid; i < 128 * 32; i += 256) {
    const int row = i >> 5;
    const int c4 = (i & 31) << 2;
    const int gr = r0 + row;
    float4 v = make_float4(0.f, 0.f, 0.f, 0.f);
    if (gr < n) v = *(const float4*)(aggr + (size_t)gr * D_DIM + c4);
    *(float4*)(As + row * 132 + c4) = v;
  }
#endif
#if HAVE_TDM
  if (tid < 32) __builtin_amdgcn_s_wait_tensorcnt(0);
#endif
  __syncthreads();

  const int wv = tid >> 5;
  const int lane = tid & 31;
  const int half = lane >> 4;
  const int ln = lane & 15;
  const float* ap = As + (wv * 16 + ln) * 132 + 2 * half;

  for (int tn = 0; tn < 16; tn += 2) {
    const float* bp0 = Bs + (tn * 16 + ln) * 132 + 2 * half;
    const float* bp1 = Bs + ((tn + 1) * 16 + ln) * 132 + 2 * half;
    v8f c0 = {}, c1 = {};
#pragma unroll
    for (int k = 0; k < 32; ++k) {
      const v2f av = *(const v2f*)(ap + 4 * k);
      const v2f b0 = *(const v2f*)(bp0 + 4 * k);
      const v2f b1v = *(const v2f*)(bp1 + 4 * k);
      c0 = __builtin_amdgcn_wmma_f32_16x16x4_f32(false, av, false, b0,
                                                 (short)0, c0, false, false);
      c1 = __builtin_amdgcn_wmma_f32_16x16x4_f32(false, av, false, b1v,
                                                 (short)0, c1, false, false);
    }
    const int col0 = tn * 16 + ln;
    const int col1 = col0 + 16;
    const float bias0 = b1[col0];
    const float bias1 = b1[col1];
#pragma unroll
    for (int v = 0; v < 8; ++v) {
      const int row = r0 + wv * 16 + v + 8 * half;
      if (row < n) {
        float v0 = c0[v] + bias0;
        float v1 = c1[v] + bias1;
        hid[(size_t)row * DFF_DIM + col0] = v0 > 0.f ? v0 : 0.f;
        hid[(size_t)row * DFF_DIM + col1] = v1 > 0.f ? v1 : 0.f;
      }
    }
  }
}

// ---------------------------------------------------------------------------
// Kernel 6: out = hid @ W2^T + b2    K=256, 119 output cols (padded to 128)
// LDS: As[128][260] + Bs[128][260]  (stride 260 == 4 mod 64 dwords)
// ---------------------------------------------------------------------------
__global__ __launch_bounds__(256) void mlp2_kernel(
    const float* __restrict__ hid, const float* __restrict__ W2,
    const float* __restrict__ b2, float* __restrict__ out, int n) {
  extern __shared__ float lds[];
  float* As = lds;                 // [128][260]
  float* Bs = lds + 128 * 260;     // [128][260] (rows >= 119 zero)
  const int tid = threadIdx.x;
  const int r0 = blockIdx.x * TILE_ROWS;

  // W2 rows 0..118 -> Bs via TDM; rows 119..127 zero-filled by threads
#if HAVE_TDM
  if (tid < 32) {
    tdm_load_2d(W2, (unsigned)(128 * 260 * 4), 256, OUT_DIM, 256, 7, 3);
  }
  for (int i = tid; i < 9 * 64; i += 256) {
    const int row = OUT_DIM + (i >> 6);
    const int c4 = (i & 63) << 2;
    *(float4*)(Bs + row * 260 + c4) = make_float4(0.f, 0.f, 0.f, 0.f);
  }
#else
  for (int i = tid; i < 128 * 64; i += 256) {
    const int row = i >> 6;
    const int c4 = (i & 63) << 2;
    float4 v = make_float4(0.f, 0.f, 0.f, 0.f);
    if (row < OUT_DIM) v = *(const float4*)(W2 + row * DFF_DIM + c4);
    float* dst = Bs + row * 260 + c4;
    dst[0] = v.x; dst[1] = v.y; dst[2] = v.z; dst[3] = v.w;
  }
#endif
  // hid tile -> As via async b128
#if ON_DEVICE
  for (int i = tid; i < 128 * 64; i += 256) {
    const int row = i >> 6;
    const int c4 = (i & 63) << 2;
    const int gr = r0 + row;
    const unsigned loff = (unsigned)((row * 260 + c4) * 4);
    if (gr < n) {
      async_g2l_b128(hid + (size_t)gr * DFF_DIM + c4, loff);
    } else {
      *(float4*)(As + row * 260 + c4) = make_float4(0.f, 0.f, 0.f, 0.f);
    }
  }
  wait_async0();
#else
  for (int i = tid; i < 128 * 64; i += 256) {
    const int row = i >> 6;
    const int c4 = (i & 63) << 2;
    const int gr = r0 + row;
    float4 v = make_float4(0.f, 0.f, 0.f, 0.f);
    if (gr < n) v = *(const float4*)(hid + (size_t)gr * DFF_DIM + c4);
    *(float4*)(As + row * 260 + c4) = v;
  }
#endif
#if HAVE_TDM
  if (tid < 32) __builtin_amdgcn_s_wait_tensorcnt(0);
#endif
  __syncthreads();

  const int wv = tid >> 5;
  const int lane = tid & 31;
  const int half = lane >> 4;
  const int ln = lane & 15;
  const float* ap = As + (wv * 16 + ln) * 260 + 2 * half;

  for (int tn = 0; tn < 8; tn += 2) {
    const float* bp0 = Bs + (tn * 16 + ln) * 260 + 2 * half;
    const float* bp1 = Bs + ((tn + 1) * 16 + ln) * 260 + 2 * half;
    v8f c0 = {}, c1 = {};
#pragma unroll
    for (int k = 0; k < 64; ++k) {
      const v2f av = *(const v2f*)(ap + 4 * k);
      const v2f b0 = *(const v2f*)(bp0 + 4 * k);
      const v2f b1v = *(const v2f*)(bp1 + 4 * k);
      c0 = __builtin_amdgcn_wmma_f32_16x16x4_f32(false, av, false, b0,
                                                 (short)0, c0, false, false);
      c1 = __builtin_amdgcn_wmma_f32_16x16x4_f32(false, av, false, b1v,
                                                 (short)0, c1, false, false);
    }
    const int col0 = tn * 16 + ln;
    const int col1 = col0 + 16;
    const float bias0 = col0 < OUT_DIM ? b2[col0] : 0.f;
    const float bias1 = col1 < OUT_DIM ? b2[col1] : 0.f;
#pragma unroll
    for (int v = 0; v < 8; ++v) {
      const int row = r0 + wv * 16 + v + 8 * half;
      if (row < n) {
        if (col0 < OUT_DIM) out[(size_t)row * OUT_DIM + col0] = c0[v] + bias0;
        if (col1 < OUT_DIM) out[(size_t)row * OUT_DIM + col1] = c1[v] + bias1;
      }
    }
  }
}

// ---------------------------------------------------------------------------
extern "C" void kernel_launch(void* const* d_in, const int* in_sizes, int n_in,
                              void* d_out, int out_size, void* d_ws, size_t ws_size,
                              hipStream_t stream) {
  const float* x          = (const float*)d_in[0];
  const int*   edge_index = (const int*)d_in[1];
  const int*   edge_attr  = (const int*)d_in[2];
  const int*   masked_idx = (const int*)d_in[3];
  const float* prelu_a    = (const float*)d_in[4];
  const float* Wenc       = (const float*)d_in[5];
  const float* emb1       = (const float*)d_in[6];
  const float* emb2       = (const float*)d_in[7];
  const float* W1         = (const float*)d_in[8];
  const float* b1         = (const float*)d_in[9];
  const float* W2         = (const float*)d_in[10];
  const float* b2         = (const float*)d_in[11];
  float* out = (float*)d_out;

  const int n     = in_sizes[0] / D_DIM;   // 50000
  const int ne    = in_sizes[1] / 2;       // 800000
  const int nmask = in_sizes[3];           // 7500

  float* h    = (float*)d_ws;
  float* aggr = h + (size_t)n * D_DIM;
  float* hid  = aggr + (size_t)n * D_DIM;

  const int nblk = (n + TILE_ROWS - 1) / TILE_ROWS;

  const size_t lds_enc  = (size_t)(2 * 128 * 132) * sizeof(float);       // 135 KB
  const size_t lds_mlp1 = (size_t)((128 + 256) * 132) * sizeof(float);   // 203 KB
  const size_t lds_mlp2 = (size_t)(2 * 128 * 260) * sizeof(float);       // 266 KB

  encode_kernel<<<nblk, 256, lds_enc, stream>>>(x, Wenc, prelu_a, h, n);
  mask_kernel<<<(nmask * 32 + 255) / 256, 256, 0, stream>>>(masked_idx, h, nmask);
  selfloop_kernel<<<(n * 32 + 255) / 256, 256, 0, stream>>>(h, emb1, emb2, aggr, n);
  scatter_kernel<<<(ne * 32 + 255) / 256, 256, 0, stream>>>(h, edge_index, edge_attr,
                                                            emb1, emb2, aggr, ne);
  mlp1_kernel<<<nblk, 256, lds_mlp1, stream>>>(aggr, W1, b1, hid, n);
  mlp2_kernel<<<nblk, 256, lds_mlp2, stream>>>(hid, W2, b2, out, n);
}